// StealLoss_42674795053124
// MI455X (gfx1250) — compile-verified
//
#include <hip/hip_runtime.h>
#include <math.h>

typedef __attribute__((ext_vector_type(2))) float v2f;
typedef __attribute__((ext_vector_type(8))) float v8f;

#define H 2048
#define W 2048
#define NPIX (H * W)
#define TAU_INV 10.0f
#define EPSF 1e-6f

// ---------------------------------------------------------------------------
// Async global->LDS staging (CDNA5 GLOBAL_LOAD_ASYNC_TO_LDS_B32, ASYNCcnt).
// Toolchain declares the builtin with (global int*, lds int*, imm, imm).
// ---------------------------------------------------------------------------
#if defined(__gfx1250__) && __has_builtin(__builtin_amdgcn_global_load_async_to_lds_b32)
#define HAVE_ASYNC_LDS 1
#endif

__device__ __forceinline__ void stage_word(const float* __restrict__ src,
                                           float* dst_lds) {
#ifdef HAVE_ASYNC_LDS
    float* s = const_cast<float*>(src);
    __builtin_amdgcn_global_load_async_to_lds_b32(
        (__attribute__((address_space(1))) int*)s,
        (__attribute__((address_space(3))) int*)dst_lds,
        /*offset=*/0, /*cpol=*/0);
#else
    *dst_lds = *src;
#endif
}

__device__ __forceinline__ void wait_async_stage() {
#ifdef HAVE_ASYNC_LDS
#if __has_builtin(__builtin_amdgcn_s_wait_asynccnt)
    __builtin_amdgcn_s_wait_asynccnt(0);
#else
    asm volatile("s_wait_asynccnt 0x0" ::: "memory");
#endif
#endif
}

// Gaussian 5-tap, sigma=2, normalized (matches _gauss_kernel5).
// Constant-index accessor -> folds to float immediates under full unroll.
__device__ __forceinline__ float gw(int k) {
    return (k == 0) ? 0.15246914f
         : (k == 1) ? 0.22184131f
         : (k == 2) ? 0.25137913f
         : (k == 3) ? 0.22184131f
                    : 0.15246914f;
}
// Band-matrix weight g[d] for runtime d (0 outside [0,4]); pure VALU selects,
// no memory traffic (g is symmetric: g0==g4, g1==g3).
__device__ __forceinline__ float gband(int d) {
    float r = 0.0f;
    r = (d == 0 || d == 4) ? 0.15246914f : r;
    r = (d == 1 || d == 3) ? 0.22184131f : r;
    r = (d == 2)           ? 0.25137913f : r;
    return r;
}

__device__ __forceinline__ int reflect_idx(int i, int n) {
    if (i < 0) i = -i;            // reflect (no edge repeat); halo<=2, one pass ok
    if (i >= n) i = 2 * n - 2 - i;
    return i;
}
__device__ __forceinline__ int clampi(int i, int n) {
    return i < 0 ? 0 : (i >= n - 1 ? n - 1 : i);
}

// ---------------------------------------------------------------------------
// Kernel 1: 5x5 separable Gaussian blur, reflect padding.
// 128 threads = 4 waves per block; block computes a 32x32 macro-tile
// (36x36 staged input via async LDS loads, 1.27x halo overhead). Row conv on
// VALU into LDS, column conv per wave as out(16x16) = Wcol(16x20) x T1(20x16)
// via 5 chained V_WMMA_F32_16X16X4_F32 with pre-gathered fragments.
// ---------------------------------------------------------------------------
__global__ __launch_bounds__(128) void steal_blur_wmma(const float* __restrict__ lab,
                                                       float* __restrict__ blur) {
    __shared__ float tIn[36][36];   // 32x32 tile + halo 2 (reflect-padded input)
    __shared__ float tMid[36][32];  // row-convolved intermediate

    const int tid = threadIdx.x;    // 0..127
    const int lane = tid & 31;
    const int wave = tid >> 5;      // 0..3 -> 2x2 arrangement of 16x16 tiles
    const int wr = (wave >> 1) * 16;
    const int wc = (wave & 1) * 16;
    const int c0 = blockIdx.x * 32;
    const int r0 = blockIdx.y * 32;

    // Stage 36x36 input macro-tile (1296 elems / 128 threads), cache->LDS async
    for (int t = tid; t < 36 * 36; t += 128) {
        int i = t / 36, j = t - i * 36;
        stage_word(&lab[(size_t)reflect_idx(r0 - 2 + i, H) * W +
                        reflect_idx(c0 - 2 + j, W)],
                   &tIn[i][j]);
    }
    wait_async_stage();
    __syncthreads();

    // Row (x) convolution: tMid[i][c] = sum_k g[k] * tIn[i][c+k]  (36x32)
    for (int t = tid; t < 36 * 32; t += 128) {
        int i = t >> 5, c = t & 31;
        float s = 0.0f;
#pragma unroll
        for (int k = 0; k < 5; ++k) s += gw(k) * tIn[i][c + k];
        tMid[i][c] = s;
    }
    __syncthreads();

    // Column (y) convolution on the matrix pipe.
    // Chunk q: A[m][k] = Wcol[m][4q+k] (Wcol[m][j] = g[j-m], band),
    //          B[k][n] = tMid[wr + 4q + k][wc + n].
    const int n = lane & 15;        // B/D column
    const int m = lane & 15;        // A row
    const int half = lane >> 4;     // K pair select {0,1} vs {2,3}

    v2f afr[5], bfr[5];
#pragma unroll
    for (int q = 0; q < 5; ++q) {
        const int jb = 4 * q + 2 * half;
        afr[q].x = gband(jb - m);
        afr[q].y = gband(jb + 1 - m);
        bfr[q].x = tMid[wr + jb][wc + n];
        bfr[q].y = tMid[wr + jb + 1][wc + n];
    }

    v8f acc = {};
#pragma unroll
    for (int q = 0; q < 5; ++q)
        acc = __builtin_amdgcn_wmma_f32_16x16x4_f32(
            /*neg_a=*/false, afr[q], /*neg_b=*/false, bfr[q],
            /*c_mod=*/(short)0, acc, /*reuse_a=*/false, /*reuse_b=*/false);

    // D layout: VGPR i -> row (i + 8*half), col = lane&15
    const int mbase = half * 8;
#pragma unroll
    for (int i = 0; i < 8; ++i)
        blur[(size_t)(r0 + wr + mbase + i) * W + (c0 + wc + n)] = acc[i];
}

// ---------------------------------------------------------------------------
// Kernel 2: second-order Sobel gradients (replicate pad at BOTH stages) ->
// normal angle -> direction section index (0..3, 4 = no section).
// 16x16 tile per block; gx/gy staged in LDS on an 18x18 halo-1 region.
// ---------------------------------------------------------------------------
__global__ __launch_bounds__(256) void steal_dir(const float* __restrict__ B,
                                                 int* __restrict__ dir) {
    __shared__ float sB[20][21];    // blur tile, halo 2, clamped
    __shared__ float sGX[18][19];   // first-order sobel-x, halo 1
    __shared__ float sGY[18][19];

    const int tx = threadIdx.x, ty = threadIdx.y;
    const int tid = ty * 16 + tx;
    const int c0 = blockIdx.x * 16;
    const int r0 = blockIdx.y * 16;

    for (int t = tid; t < 400; t += 256) {
        int i = t / 20, j = t - i * 20;
        stage_word(&B[(size_t)clampi(r0 - 2 + i, H) * W + clampi(c0 - 2 + j, W)],
                   &sB[i][j]);
    }
    wait_async_stage();
    __syncthreads();

    // gx/gy over 18x18 (abs rows r0-1..r0+16); center in sB at (u+1, v+1)
    for (int t = tid; t < 324; t += 256) {
        int u = t / 18, v = t - u * 18;
        int i = u + 1, j = v + 1;
        float a = sB[i - 1][j - 1], b2 = sB[i - 1][j], c = sB[i - 1][j + 1];
        float d = sB[i][j - 1],                        f = sB[i][j + 1];
        float g = sB[i + 1][j - 1], h2 = sB[i + 1][j], k = sB[i + 1][j + 1];
        sGX[u][v] = (-a + c - 2.0f * d + 2.0f * f - g + k) * 0.125f;
        sGY[u][v] = (-a - 2.0f * b2 - c + g + 2.0f * h2 + k) * 0.125f;
    }
    __syncthreads();

    const int r = r0 + ty, c = c0 + tx;
    // outer stage clamps coordinates to the image, then maps into GX/GY tile
    int uu[3], vv[3];
#pragma unroll
    for (int d = -1; d <= 1; ++d) {
        uu[d + 1] = clampi(r + d, H) - (r0 - 1);
        vv[d + 1] = clampi(c + d, W) - (c0 - 1);
    }
    float gxx = (-sGX[uu[0]][vv[0]] + sGX[uu[0]][vv[2]]
                 - 2.0f * sGX[uu[1]][vv[0]] + 2.0f * sGX[uu[1]][vv[2]]
                 - sGX[uu[2]][vv[0]] + sGX[uu[2]][vv[2]]) * 0.125f;
    float gxy = (-sGY[uu[0]][vv[0]] + sGY[uu[0]][vv[2]]
                 - 2.0f * sGY[uu[1]][vv[0]] + 2.0f * sGY[uu[1]][vv[2]]
                 - sGY[uu[2]][vv[0]] + sGY[uu[2]][vv[2]]) * 0.125f;
    float gyy = (-sGY[uu[0]][vv[0]] - 2.0f * sGY[uu[0]][vv[1]] - sGY[uu[0]][vv[2]]
                 + sGY[uu[2]][vv[0]] + 2.0f * sGY[uu[2]][vv[1]] + sGY[uu[2]][vv[2]]) * 0.125f;

    float t = gxy + EPSF;
    float s = (t > 0.0f) ? -1.0f : ((t < 0.0f) ? 1.0f : 0.0f);
    float a = atanf(gyy * s / (gxx + EPSF));

    const float PI8 = 0.39269908169872414f;   // pi/8
    const float PI38 = 1.1780972450961724f;   // 3pi/8
    const float PI2 = 1.5707963267948966f;    // pi/2
    int d = 4;
    if (a >= -PI8 && a < PI8) d = 0;                                    // horiz
    else if ((a >= PI38 && a < PI2) || (a >= -PI2 && a < -PI38)) d = 1; // vert
    else if (a >= PI8 && a < PI38) d = 2;                               // lead diag
    else if (a >= -PI38 && a < -PI8) d = 3;                             // cnt diag
    dir[(size_t)r * W + c] = d;
}

// ---------------------------------------------------------------------------
// Kernel 3: directional softmax-style loss + block reduction + atomic sum.
// loss = pred/tau - log(resp_d + eps), resp_d = 5-tap zero-padded sum of
// exp(pred/tau) along the selected direction.
// ---------------------------------------------------------------------------
__global__ __launch_bounds__(256) void steal_loss(const float* __restrict__ pred,
                                                  const int* __restrict__ dir,
                                                  float* __restrict__ accum) {
    __shared__ float red[256];
    const int tid = threadIdx.x;
    const size_t idx = (size_t)blockIdx.x * 256 + tid;
    const int r = (int)(idx >> 11);        // /2048
    const int c = (int)(idx & 2047);

    // Warm the cache for the row below (vertical/diagonal taps) -> global_prefetch_b8
    if (r + 1 < H) __builtin_prefetch(&pred[idx + W], 0, 0);

    const int DR[4] = {0, 1, 1, 1};
    const int DC[4] = {1, 0, -1, 1};

    float loss = 0.0f;
    int d = dir[idx];
    if (d < 4) {
        float resp = 0.0f;
#pragma unroll
        for (int t = -2; t <= 2; ++t) {
            int rr = r + t * DR[d];
            int cc = c + t * DC[d];
            if (rr >= 0 && rr < H && cc >= 0 && cc < W)
                resp += __expf(pred[(size_t)rr * W + cc] * TAU_INV);
        }
        loss = pred[idx] * TAU_INV - __logf(resp + EPSF);
    }

    red[tid] = loss;
    __syncthreads();
#pragma unroll
    for (int s = 128; s > 0; s >>= 1) {
        if (tid < s) red[tid] += red[tid + s];
        __syncthreads();
    }
    if (tid == 0) atomicAdd(accum, red[0]);
}

__global__ void steal_zero(float* accum) { *accum = 0.0f; }
__global__ void steal_fin(const float* accum, float* out) {
    out[0] = -accum[0] * (1.0f / (float)NPIX);
}

// ---------------------------------------------------------------------------
extern "C" void kernel_launch(void* const* d_in, const int* in_sizes, int n_in,
                              void* d_out, int out_size, void* d_ws, size_t ws_size,
                              hipStream_t stream) {
    (void)in_sizes; (void)n_in; (void)out_size; (void)ws_size;
    const float* pred = (const float*)d_in[0];   // (1,1,2048,2048) f32
    const float* lab  = (const float*)d_in[1];   // (1,2048,2048)   f32
    float* out = (float*)d_out;                  // scalar f32

    float* B    = (float*)d_ws;                                   // N floats
    int*   dir  = (int*)((char*)d_ws + (size_t)NPIX * 4);         // N ints
    float* acc  = (float*)((char*)d_ws + (size_t)2 * NPIX * 4);   // 1 float

    steal_zero<<<1, 1, 0, stream>>>(acc);
    steal_blur_wmma<<<dim3(W / 32, H / 32), 128, 0, stream>>>(lab, B);
    steal_dir<<<dim3(W / 16, H / 16), dim3(16, 16), 0, stream>>>(B, dir);
    steal_loss<<<NPIX / 256, 256, 0, stream>>>(pred, dir, acc);
    steal_fin<<<1, 1, 0, stream>>>(acc, out);
}